// ImageSectionRNNCell_34900904247540
// MI455X (gfx1250) — compile-verified
//
#include <hip/hip_runtime.h>
#include <hip/hip_bf16.h>
#include <math.h>

// ---------------------------------------------------------------------------
// ImageSectionRNNCell for MI455X (gfx1250), fp32 via V_WMMA_F32_16X16X4_F32.
// B=64, U=256, grid 16x16, conv stack -> 432-flat -> dense -> LSTM -> nsc.
// ---------------------------------------------------------------------------

typedef __attribute__((ext_vector_type(2))) float v2f;
typedef __attribute__((ext_vector_type(8))) float v8f;

#define BATCH 64
#define UNITS 256
#define KFLAT 432   // 12*12*3

// fp32 WMMA: D(16x16,f32) = A(16x4,f32) x B(4x16,f32) + C
__device__ __forceinline__ v8f wmma4_f32(v2f a, v2f b, v8f c) {
    return __builtin_amdgcn_wmma_f32_16x16x4_f32(
        /*neg_a=*/false, a, /*neg_b=*/false, b,
        /*c_mod=*/(short)0, c, /*reuse_a=*/false, /*reuse_b=*/false);
}

__device__ __forceinline__ float sigmoidf_(float x) {
    return 1.0f / (1.0f + __expf(-x));
}

// ---------------------------------------------------------------------------
// Kernel 1: bilinear section sampling + conv1(relu) + conv2(relu) -> Xf(64x432)
// One block per batch element, 256 threads.
// ---------------------------------------------------------------------------
__global__ __launch_bounds__(256)
void frontend_kernel(const float* __restrict__ image,
                     const float* __restrict__ section,
                     const float* __restrict__ c1k, const float* __restrict__ c1b,
                     const float* __restrict__ c2k, const float* __restrict__ c2b,
                     float* __restrict__ Xf) {
    __shared__ float grid[16][16][3];
    __shared__ float t1[14][14][3];

    const int b   = blockIdx.x;
    const int tid = threadIdx.x;

    const float s0 = section[b * 3 + 0];
    const float s1 = section[b * 3 + 1];
    const float s2 = section[b * 3 + 2];   // broadcast into both dims (stops = starts + s2)

    // --- bilinear sample: thread (i,j) computes 3 channels -------------------
    {
        const int i = tid >> 4;
        const int j = tid & 15;
        const float t  = (float)i * (1.0f / 15.0f);
        const float al = (float)j * (1.0f / 15.0f);
        const float p0 = s0 + t  * s2;
        const float p1 = s1 + al * s2;
        float fy = fminf(fmaxf(p0 * 511.0f, 0.0f), 511.0f);
        float fx = fminf(fmaxf(p1 * 511.0f, 0.0f), 511.0f);
        int y0 = (int)floorf(fy);
        int x0 = (int)floorf(fx);
        int y1 = min(y0 + 1, 511);
        int x1 = min(x0 + 1, 511);
        float wy = fy - (float)y0;
        float wx = fx - (float)x0;
        const float* img = image + (size_t)b * 512 * 512 * 3;
        #pragma unroll
        for (int c = 0; c < 3; ++c) {
            float v00 = img[((size_t)y0 * 512 + x0) * 3 + c];
            float v01 = img[((size_t)y0 * 512 + x1) * 3 + c];
            float v10 = img[((size_t)y1 * 512 + x0) * 3 + c];
            float v11 = img[((size_t)y1 * 512 + x1) * 3 + c];
            float top = v00 * (1.0f - wx) + v01 * wx;
            float bot = v10 * (1.0f - wx) + v11 * wx;
            grid[i][j][c] = top * (1.0f - wy) + bot * wy;
        }
    }
    __syncthreads();

    // --- conv1: 3x3 VALID, 3->3 ch, relu -> t1[14][14][3] --------------------
    for (int idx = tid; idx < 14 * 14 * 3; idx += 256) {
        const int o = idx % 3;
        const int j = (idx / 3) % 14;
        const int i = idx / (3 * 14);
        float acc = c1b[o];
        #pragma unroll
        for (int di = 0; di < 3; ++di)
            #pragma unroll
            for (int dj = 0; dj < 3; ++dj)
                #pragma unroll
                for (int ci = 0; ci < 3; ++ci)
                    acc += grid[i + di][j + dj][ci] * c1k[((di * 3 + dj) * 3 + ci) * 3 + o];
        t1[i][j][o] = fmaxf(acc, 0.0f);
    }
    __syncthreads();

    // --- conv2: 3x3 VALID, 3->3 ch, relu -> Xf[b][432] (NHWC flatten) --------
    for (int idx = tid; idx < 12 * 12 * 3; idx += 256) {
        const int o = idx % 3;
        const int j = (idx / 3) % 12;
        const int i = idx / (3 * 12);
        float acc = c2b[o];
        #pragma unroll
        for (int di = 0; di < 3; ++di)
            #pragma unroll
            for (int dj = 0; dj < 3; ++dj)
                #pragma unroll
                for (int ci = 0; ci < 3; ++ci)
                    acc += t1[i + di][j + dj][ci] * c2k[((di * 3 + dj) * 3 + ci) * 3 + o];
        Xf[b * KFLAT + (i * 12 + j) * 3 + o] = fmaxf(acc, 0.0f);
    }
}

// ---------------------------------------------------------------------------
// Generic fp32 WMMA GEMM: C(MxN) = act(A(MxK) @ B(KxN) + bias), one wave per
// 16x16 tile. K % 4 == 0, M,N % 16 == 0. relu: 1 -> max(0,x).
// ---------------------------------------------------------------------------
__global__ __launch_bounds__(256)
void gemm_wmma_kernel(const float* __restrict__ A, const float* __restrict__ B,
                      const float* __restrict__ bias, float* __restrict__ C,
                      int M, int N, int K, int relu) {
    const int wave = blockIdx.x * (blockDim.x >> 5) + (threadIdx.x >> 5);
    const int lane = threadIdx.x & 31;
    const int tilesN = N >> 4;
    const int tm = (wave / tilesN) << 4;
    const int tn = (wave % tilesN) << 4;
    if (tm >= M) return;                       // wave-uniform: EXEC stays all-1s

    const int mrow = lane & 15;                // A row / C column index
    const int half = lane >> 4;                // 0: K {0,1}, 1: K {2,3}
    const int kk   = half << 1;
    const int ncol = lane & 15;                // B/C column

    const float* Ar = A + (size_t)(tm + mrow) * K;
    const float* Bc = B + tn + ncol;

    v8f acc = {};
    for (int k = 0; k < K; k += 4) {
        v2f av, bv;
        av.x = Ar[k + kk];
        av.y = Ar[k + kk + 1];
        bv.x = Bc[(size_t)(k + kk) * N];
        bv.y = Bc[(size_t)(k + kk + 1) * N];
        acc = wmma4_f32(av, bv, acc);
    }

    const float bcol = bias[tn + ncol];
    #pragma unroll
    for (int r = 0; r < 8; ++r) {
        const int row = tm + r + (half << 3);
        float v = acc[r] + bcol;
        if (relu) v = fmaxf(v, 0.0f);
        C[(size_t)row * N + tn + ncol] = v;
    }
}

// ---------------------------------------------------------------------------
// LSTM pre-activation: Z(64x1024) = X1 @ Wk + H0 @ Wr + bias. Dual WMMA/K-step.
// ---------------------------------------------------------------------------
__global__ __launch_bounds__(256)
void lstm_gemm_kernel(const float* __restrict__ X1, const float* __restrict__ H0,
                      const float* __restrict__ Wk, const float* __restrict__ Wr,
                      const float* __restrict__ bias, float* __restrict__ Z) {
    const int N = 4 * UNITS;                   // 1024
    const int K = UNITS;                       // 256
    const int wave = blockIdx.x * (blockDim.x >> 5) + (threadIdx.x >> 5);
    const int lane = threadIdx.x & 31;
    const int tilesN = N >> 4;                 // 64
    const int tm = (wave / tilesN) << 4;
    const int tn = (wave % tilesN) << 4;

    const int mrow = lane & 15;
    const int half = lane >> 4;
    const int kk   = half << 1;
    const int ncol = lane & 15;

    const float* x1r = X1 + (size_t)(tm + mrow) * K;
    const float* h0r = H0 + (size_t)(tm + mrow) * K;
    const float* wkc = Wk + tn + ncol;
    const float* wrc = Wr + tn + ncol;

    v8f acc = {};
    for (int k = 0; k < K; k += 4) {
        v2f av, bv;
        av.x = x1r[k + kk];          av.y = x1r[k + kk + 1];
        bv.x = wkc[(size_t)(k + kk) * N]; bv.y = wkc[(size_t)(k + kk + 1) * N];
        acc = wmma4_f32(av, bv, acc);
        av.x = h0r[k + kk];          av.y = h0r[k + kk + 1];
        bv.x = wrc[(size_t)(k + kk) * N]; bv.y = wrc[(size_t)(k + kk + 1) * N];
        acc = wmma4_f32(av, bv, acc);
    }

    const float bcol = bias[tn + ncol];
    #pragma unroll
    for (int r = 0; r < 8; ++r) {
        const int row = tm + r + (half << 3);
        Z[(size_t)row * N + tn + ncol] = acc[r] + bcol;
    }
}

// ---------------------------------------------------------------------------
// LSTM cell elementwise: gates from Z, produce h_new / c_new into d_out.
// ---------------------------------------------------------------------------
__global__ __launch_bounds__(256)
void lstm_cell_kernel(const float* __restrict__ Z, const float* __restrict__ C0,
                      float* __restrict__ h_new, float* __restrict__ c_new) {
    const int tid = blockIdx.x * blockDim.x + threadIdx.x;   // 64*256 threads
    const int b = tid >> 8;
    const int u = tid & 255;
    const float* zr = Z + (size_t)b * (4 * UNITS);
    const float zi = zr[u];
    const float zf = zr[UNITS + u];
    const float zg = zr[2 * UNITS + u];
    const float zo = zr[3 * UNITS + u];
    const float c = sigmoidf_(zf) * C0[tid] + sigmoidf_(zi) * tanhf(zg);
    const float h = sigmoidf_(zo) * tanhf(c);
    c_new[tid] = c;
    h_new[tid] = h;
}

// ---------------------------------------------------------------------------
// next_section = sigmoid(NSC(64x256) @ w2(256x3) + b2) -> 64x3 (too skinny for WMMA)
// ---------------------------------------------------------------------------
__global__ __launch_bounds__(192)
void next_section_kernel(const float* __restrict__ NSC,
                         const float* __restrict__ w2, const float* __restrict__ b2,
                         float* __restrict__ out) {
    const int tid = threadIdx.x;               // 192 = 64*3
    const int b = tid / 3;
    const int j = tid % 3;
    const float* row = NSC + (size_t)b * UNITS;
    float acc = b2[j];
    for (int k = 0; k < UNITS; ++k)
        acc += row[k] * w2[k * 3 + j];
    out[tid] = sigmoidf_(acc);
}

// ---------------------------------------------------------------------------
// Host-side launch
// ---------------------------------------------------------------------------
extern "C" void kernel_launch(void* const* d_in, const int* in_sizes, int n_in,
                              void* d_out, int out_size, void* d_ws, size_t ws_size,
                              hipStream_t stream) {
    const float* image   = (const float*)d_in[0];
    const float* section = (const float*)d_in[1];
    const float* h0      = (const float*)d_in[2];
    const float* c0      = (const float*)d_in[3];
    const float* conv1_k = (const float*)d_in[4];
    const float* conv1_b = (const float*)d_in[5];
    const float* conv2_k = (const float*)d_in[6];
    const float* conv2_b = (const float*)d_in[7];
    const float* dense_w = (const float*)d_in[8];
    const float* dense_b = (const float*)d_in[9];
    const float* lstm_k  = (const float*)d_in[10];
    const float* lstm_rk = (const float*)d_in[11];
    const float* lstm_b  = (const float*)d_in[12];
    const float* nsc_w1  = (const float*)d_in[13];
    const float* nsc_b1  = (const float*)d_in[14];
    const float* nsc_w2  = (const float*)d_in[15];
    const float* nsc_b2  = (const float*)d_in[16];

    float* out   = (float*)d_out;
    float* h_new = out;                         // 64*256
    float* c_new = out + BATCH * UNITS;         // 64*256
    float* nsec  = out + 2 * BATCH * UNITS;     // 64*3

    // workspace layout (floats)
    float* ws  = (float*)d_ws;
    float* Xf  = ws;                            // 64*432   = 27648
    float* X1  = Xf + BATCH * KFLAT;            // 64*256   = 16384
    float* Z   = X1 + BATCH * UNITS;            // 64*1024  = 65536
    float* NSC = Z  + BATCH * 4 * UNITS;        // 64*256   = 16384

    // 1) sample + convs -> Xf
    frontend_kernel<<<BATCH, 256, 0, stream>>>(image, section, conv1_k, conv1_b,
                                               conv2_k, conv2_b, Xf);

    // 2) dense: X1 = relu(Xf @ dense_w + dense_b)   M=64 N=256 K=432 -> 64 tiles
    gemm_wmma_kernel<<<8, 256, 0, stream>>>(Xf, dense_w, dense_b, X1,
                                            BATCH, UNITS, KFLAT, 1);

    // 3) LSTM pre-activation: Z = X1 @ lstm_k + h0 @ lstm_rk + lstm_b -> 256 tiles
    lstm_gemm_kernel<<<32, 256, 0, stream>>>(X1, h0, lstm_k, lstm_rk, lstm_b, Z);

    // 4) gates -> h_new, c_new
    lstm_cell_kernel<<<(BATCH * UNITS) / 256, 256, 0, stream>>>(Z, c0, h_new, c_new);

    // 5) nsc = relu(h_new @ nsc_w1 + nsc_b1)        M=64 N=256 K=256 -> 64 tiles
    gemm_wmma_kernel<<<8, 256, 0, stream>>>(h_new, nsc_w1, nsc_b1, NSC,
                                            BATCH, UNITS, UNITS, 1);

    // 6) next_section = sigmoid(nsc @ nsc_w2 + nsc_b2)
    next_section_kernel<<<1, 192, 0, stream>>>(NSC, nsc_w2, nsc_b2, nsec);

    (void)in_sizes; (void)n_in; (void)out_size; (void)ws_size;
}